// CrossEntropyWithProbs_52261162058056
// MI455X (gfx1250) — compile-verified
//
#include <hip/hip_runtime.h>
#include <cstdint>

// ---------------------------------------------------------------------------
// CrossEntropyWithProbs (soft labels + class weights), N=2^21 rows, C=32.
//   loss_n = (m_n + log S_n) * A_n - B_n
//     m_n = max_c x, S_n = sum_c exp(x-m), A_n = sum_c t*w, B_n = sum_c t*w*x
//   out  = mean_n loss_n
// Memory-bound: streams 512 MB once -> ~22 us at 23.3 TB/s. Strategy:
//   * async global->LDS B128 copies (ASYNCcnt), double-buffered per block
//   * LDS rows padded to stride 36 floats (144 B, 16B-aligned) -> minimal
//     bank conflicts for per-thread ds_load_b128 row reads
//   * thread-per-row math; exp/log on trans pipe
//   * wave reduction via V_WMMA_F32_16X16X4_F32 (all-ones B => row sums)
//   * block sum -> f64 global atomicAdd into d_ws; finalize kernel -> mean
// ---------------------------------------------------------------------------

#define CE_C      32
#define CE_TILE   256
#define CE_BLOCK  256
#define CE_RSTR   36          // padded LDS row stride in floats
#define CE_NBLK   2048

typedef __attribute__((ext_vector_type(2))) float v2f;
typedef __attribute__((ext_vector_type(4))) float v4f;
typedef __attribute__((ext_vector_type(8))) float v8f;
typedef __attribute__((ext_vector_type(4))) int   v4i;

#define AS1 __attribute__((address_space(1)))
#define AS3 __attribute__((address_space(3)))

#define HAS_ASYNC __has_builtin(__builtin_amdgcn_global_load_async_to_lds_b128)
#define HAS_WAITA __has_builtin(__builtin_amdgcn_s_wait_asynccnt)
#define HAS_WMMA4 __has_builtin(__builtin_amdgcn_wmma_f32_16x16x4_f32)

__device__ __forceinline__ void ce_async_b128(const float* g, float* l) {
#if HAS_ASYNC
  // (global v4i* src, local v4i* dst, imm offset, imm cpol)
  __builtin_amdgcn_global_load_async_to_lds_b128(
      (AS1 v4i*)(uintptr_t)g, (AS3 v4i*)l, 0, 0);
#else
  *(v4f*)l = *(const v4f*)g;   // fallback: b128 load + ds_store_b128
#endif
}

__device__ __forceinline__ void ce_wait_async(bool keep16) {
#if HAS_ASYNC
#if HAS_WAITA
  if (keep16) __builtin_amdgcn_s_wait_asynccnt(16);
  else        __builtin_amdgcn_s_wait_asynccnt(0);
#else
  if (keep16) asm volatile("s_wait_asynccnt 16" ::: "memory");
  else        asm volatile("s_wait_asynccnt 0"  ::: "memory");
#endif
#else
  (void)keep16;
#endif
}

__global__ __launch_bounds__(CE_BLOCK)
void ce_main(const float* __restrict__ x, const float* __restrict__ t,
             const float* __restrict__ w, double* __restrict__ partial, int n) {
  __shared__ __align__(16) float xs[2][CE_TILE * CE_RSTR];
  __shared__ __align__(16) float ts[2][CE_TILE * CE_RSTR];
  __shared__ float wsh[CE_C];
  __shared__ float sred[CE_BLOCK / 32];

  const int tid = threadIdx.x;
  if (tid < CE_C) wsh[tid] = w[tid];

  const int ntiles = (n + CE_TILE - 1) / CE_TILE;

  // 8 B128 transfers per thread per array per tile (16 async instrs / wave)
  auto copy_tile = [&](int buf, int tileIdx) {
#pragma unroll
    for (int i = 0; i < (CE_TILE * CE_C) / (CE_BLOCK * 4); ++i) {
      int idx = i * CE_BLOCK + tid;   // b128 transfer index within tile
      int row = idx >> 3;
      int cb  = idx & 7;
      int grow = tileIdx * CE_TILE + row;
      if (grow >= n) grow = n - 1;    // clamp (N % TILE == 0 in practice)
      size_t gOff = (size_t)grow * CE_C + (size_t)cb * 4;
      int    lOff = row * CE_RSTR + cb * 4;
      ce_async_b128(x + gOff, &xs[buf][lOff]);
      ce_async_b128(t + gOff, &ts[buf][lOff]);
    }
  };

  float acc = 0.f;
  const int tile0 = blockIdx.x;
  if (tile0 < ntiles) copy_tile(0, tile0);

  int it = 0;
  for (int tile = tile0; tile < ntiles; tile += (int)gridDim.x, ++it) {
    const int  cur = it & 1;
    const int  nxt = tile + (int)gridDim.x;
    const bool hasNext = nxt < ntiles;
    if (hasNext) copy_tile(cur ^ 1, nxt);
    ce_wait_async(hasNext);           // <=16 keeps next tile's copies in flight
    __syncthreads();

    const int row = tile * CE_TILE + tid;
    if (row < n) {
      const v4f* xr = (const v4f*)&xs[cur][tid * CE_RSTR];
      const v4f* tr = (const v4f*)&ts[cur][tid * CE_RSTR];
      float xv[CE_C];
#pragma unroll
      for (int i = 0; i < 8; ++i) {
        v4f v = xr[i];
#pragma unroll
        for (int j = 0; j < 4; ++j) xv[4 * i + j] = v[j];
      }
      float m = xv[0];
#pragma unroll
      for (int c = 1; c < CE_C; ++c) m = fmaxf(m, xv[c]);
      float s = 0.f, A = 0.f, B = 0.f;
#pragma unroll
      for (int i = 0; i < 8; ++i) {
        v4f tv = tr[i];
#pragma unroll
        for (int j = 0; j < 4; ++j) {
          const int c = 4 * i + j;
          const float tw = tv[j] * wsh[c];
          A += tw;
          B = fmaf(tw, xv[c], B);
          s += __expf(xv[c] - m);
        }
      }
      const float lse = m + __logf(s);
      acc += fmaf(lse, A, -B);
    }
    __syncthreads();                  // everyone done with `cur` before reuse
  }

  // ---- wave reduction via the matrix pipe ---------------------------------
  float wsum;
#if HAS_WMMA4
  {
    // A(16x4): lane m -> A[m,0..1], lane m+16 -> A[m,2..3]; put acc in K=0/K=2.
    // B(4x16) = all ones  =>  D[m][*] = acc@m + acc@(m+16).
    v2f a; a[0] = acc; a[1] = 0.f;
    v2f b; b[0] = 1.f; b[1] = 1.f;
    v8f c = {};
    v8f d = __builtin_amdgcn_wmma_f32_16x16x4_f32(
        false, a, false, b, (short)0, c, false, false);
    float p = d[0] + d[1] + d[2] + d[3] + d[4] + d[5] + d[6] + d[7];
    wsum = p + __shfl_xor(p, 16, 32); // rows 0..7 half + rows 8..15 half
  }
#else
  wsum = acc;
#pragma unroll
  for (int off = 16; off; off >>= 1) wsum += __shfl_xor(wsum, off, 32);
#endif

  if ((tid & 31) == 0) sred[tid >> 5] = wsum;
  __syncthreads();
  if (tid == 0) {
    float bsum = 0.f;
#pragma unroll
    for (int i = 0; i < CE_BLOCK / 32; ++i) bsum += sred[i];
    atomicAdd(partial, (double)bsum);   // global_atomic_add_f64
  }
}

__global__ void ce_finalize(const double* __restrict__ partial,
                            float* __restrict__ out, int n) {
  out[0] = (float)(partial[0] / (double)n);
}

extern "C" void kernel_launch(void* const* d_in, const int* in_sizes, int n_in,
                              void* d_out, int out_size, void* d_ws, size_t ws_size,
                              hipStream_t stream) {
  (void)n_in; (void)out_size; (void)ws_size;
  const float* x = (const float*)d_in[0];
  const float* t = (const float*)d_in[1];
  const float* w = (const float*)d_in[2];
  const int n = in_sizes[0] / CE_C;          // number of rows
  double* partial = (double*)d_ws;

  (void)hipMemsetAsync(partial, 0, sizeof(double), stream);  // capture-legal
  const int ntiles = (n + CE_TILE - 1) / CE_TILE;
  const int grid = (CE_NBLK < ntiles) ? CE_NBLK : ntiles;
  ce_main<<<grid, CE_BLOCK, 0, stream>>>(x, t, w, partial, n);
  ce_finalize<<<1, 1, 0, stream>>>(partial, (float*)d_out, n);
}